// TipAdatperSA_25958782337697
// MI455X (gfx1250) — compile-verified
//
#include <hip/hip_runtime.h>

typedef __attribute__((ext_vector_type(2))) float v2f;
typedef __attribute__((ext_vector_type(8))) float v8f;

#define BETA_F 5.5f

// Problem constants (from reference setup_inputs)
constexpr int N_ = 128;      // batch
constexpr int C_ = 1000;     // classes / value dim
constexpr int D_ = 1024;     // feature dim
constexpr int K_ = 16;       // shots per class
constexpr int CK_ = C_ * K_; // 16000
constexpr int KCHUNK = 1600; // 16000 / 10 split-K chunks
constexpr int KSUB = 64;     // values staging sub-chunk (k rows per LDS refill)
constexpr int BSTRIDE = 33;  // padded LDS row stride (32 cols + 1) -> no bank conflicts

// ---------------------------------------------------------------------------
// Kernel 1: affinity[n, c*16+j] = dot(x[n,c,:], keys[c*16+j,:])
// One block per class c. 8 waves; wave w computes the 16x16 tile
// (rows n = 16w..16w+15, cols j = 0..15) via 256 steps of WMMA f32 16x16x4.
// keys tile staged in LDS transposed [d][j] (64 KB), conflict-free B reads.
// x streamed with NT hint (524 MB, read once -> don't pollute L2).
// ---------------------------------------------------------------------------
template <bool HAS_WS>
__global__ __launch_bounds__(256) void aff_kernel(
    const float* __restrict__ x, const float* __restrict__ keys,
    float* __restrict__ aff, float* __restrict__ expw)
{
    __shared__ float bk[D_ * 16]; // [d][j], 65536 bytes

    const int c   = blockIdx.x;
    const int tid = threadIdx.x;

    // Cooperative transpose-load of keys[c*16 + j][d] -> bk[d*16 + j].
    {
        const int r     = tid & 15;
        const int dbase = (tid >> 4) * 64;
        const float* krow = keys + (size_t)(c * K_ + r) * D_ + dbase;
        #pragma unroll
        for (int i = 0; i < 64; ++i)
            bk[(dbase + i) * 16 + r] = krow[i];
    }
    __syncthreads();

    const int wave  = tid >> 5;      // 0..7 -> M-tile
    const int lane  = tid & 31;
    const int half  = lane >> 4;     // 0: K pair {0,1}; 1: K pair {2,3}
    const int lrow  = lane & 15;     // A: row M; B/C/D: col N
    const int mbase = wave * 16;

    const float* arow = x + (size_t)(mbase + lrow) * (C_ * D_)
                          + (size_t)c * D_ + 2 * half;
    const float* bbase = &bk[(2 * half) * 16 + lrow];

    v8f acc0 = {};
    v8f acc1 = {};
    #pragma unroll 4
    for (int k4 = 0; k4 < D_; k4 += 8) {
        v2f a0 = __builtin_nontemporal_load((const v2f*)(arow + k4));
        v2f b0;
        b0.x = bbase[(k4 + 0) * 16];
        b0.y = bbase[(k4 + 1) * 16];
        acc0 = __builtin_amdgcn_wmma_f32_16x16x4_f32(
            false, a0, false, b0, (short)0, acc0, false, false);

        v2f a1 = __builtin_nontemporal_load((const v2f*)(arow + k4 + 4));
        v2f b1;
        b1.x = bbase[(k4 + 4) * 16];
        b1.y = bbase[(k4 + 5) * 16];
        acc1 = __builtin_amdgcn_wmma_f32_16x16x4_f32(
            false, a1, false, b1, (short)0, acc1, false, false);
    }

    // Epilogue: NaN->0, store affinity (NT: pure output), precompute exp.
    #pragma unroll
    for (int v = 0; v < 8; ++v) {
        const int n = mbase + v + 8 * half; // C/D layout: vgpr v, two halves
        const int j = lrow;
        float af = acc0[v] + acc1[v];
        af = (af != af) ? 0.0f : af;        // jnp.where(isnan, 0, .)
        const size_t idx = (size_t)n * CK_ + (size_t)c * K_ + j;
        __builtin_nontemporal_store(af, &aff[idx]);
        if (HAS_WS)
            expw[idx] = __expf(fmaf(BETA_F, af, -BETA_F));
    }
}

// ---------------------------------------------------------------------------
// Kernel 2: out[n, v] = sum_j expaff[n, j] * values[j, v]
// Grid (32 col-pair-tiles, 10 K-chunks) x 8 M-tile waves. Each block owns
// 32 output columns (two 16-wide WMMA N-tiles) so every A fragment feeds two
// independent WMMA chains. values sub-tiles ([64 x 32], padded stride 33)
// staged cooperatively in LDS -> each values element fetched from HBM once.
// Split-K partial sums combined with f32 atomics into pre-zeroed out.
// ---------------------------------------------------------------------------
template <bool APPLY_EXP>
__global__ __launch_bounds__(256) void out_kernel(
    const float* __restrict__ A,       // [N_, CK_] expaff (or raw affinity)
    const float* __restrict__ values,  // [CK_, C_]
    float* __restrict__ out)           // [N_, C_], pre-zeroed
{
    __shared__ float bv[KSUB * BSTRIDE]; // [k][col0..31] padded, 8448 bytes

    const int colbase = blockIdx.x * 32;  // 0..992 (cols 1000..1023 are pad)
    const int kchunk  = blockIdx.y;       // 0..9
    const int tid     = threadIdx.x;
    const int wave    = tid >> 5;
    const int lane    = tid & 31;
    const int half    = lane >> 4;
    const int lrow    = lane & 15;
    const int mbase   = wave * 16;
    const int k0      = kchunk * KCHUNK;

    // Fill assignment: thread t -> col t&31, rows (t>>5)*8 .. +7
    const int fcol  = tid & 31;
    const int fcolc = (colbase + fcol < C_) ? (colbase + fcol) : (C_ - 1);
    const int frow0 = (tid >> 5) * 8;

    const float* arow = A + (size_t)(mbase + lrow) * CK_ + k0 + 2 * half;

    v8f acc0 = {}; // columns colbase + 0..15
    v8f acc1 = {}; // columns colbase + 16..31

    for (int ks = 0; ks < KCHUNK; ks += KSUB) {
        __syncthreads(); // previous sub-chunk's compute done before overwrite
        #pragma unroll
        for (int i = 0; i < 8; ++i) {
            const int r = frow0 + i;
            bv[r * BSTRIDE + fcol] =
                values[(size_t)(k0 + ks + r) * C_ + fcolc];
        }
        __syncthreads(); // fill visible

        const float* bbase = &bv[(2 * half) * BSTRIDE + lrow];
        #pragma unroll 4
        for (int k4 = 0; k4 < KSUB; k4 += 4) {
            v2f a = *(const v2f*)(arow + ks + k4);
            if (APPLY_EXP) {
                a.x = __expf(fmaf(BETA_F, a.x, -BETA_F));
                a.y = __expf(fmaf(BETA_F, a.y, -BETA_F));
            }
            v2f b0, b1;
            b0.x = bbase[(k4 + 0) * BSTRIDE];
            b0.y = bbase[(k4 + 1) * BSTRIDE];
            b1.x = bbase[(k4 + 0) * BSTRIDE + 16];
            b1.y = bbase[(k4 + 1) * BSTRIDE + 16];
            acc0 = __builtin_amdgcn_wmma_f32_16x16x4_f32(
                false, a, false, b0, (short)0, acc0, false, false);
            acc1 = __builtin_amdgcn_wmma_f32_16x16x4_f32(
                false, a, false, b1, (short)0, acc1, false, false);
        }
    }

    const int col0 = colbase + lrow;
    const int col1 = colbase + 16 + lrow;
    #pragma unroll
    for (int v = 0; v < 8; ++v) {
        const int n = mbase + v + 8 * half;
        if (col0 < C_) atomicAdd(&out[(size_t)n * C_ + col0], acc0[v]);
        if (col1 < C_) atomicAdd(&out[(size_t)n * C_ + col1], acc1[v]);
    }
}

// ---------------------------------------------------------------------------
extern "C" void kernel_launch(void* const* d_in, const int* in_sizes, int n_in,
                              void* d_out, int out_size, void* d_ws, size_t ws_size,
                              hipStream_t stream)
{
    const float* x      = (const float*)d_in[0]; // [N, C, D]
    const float* keys   = (const float*)d_in[1]; // [K*C, D]
    const float* values = (const float*)d_in[2]; // [K*C, C]

    float* out = (float*)d_out;              // [N, C]
    float* aff = out + (size_t)N_ * C_;      // [N, C*K]

    const size_t need = (size_t)N_ * CK_ * sizeof(float);
    const bool has_ws = (d_ws != nullptr) && (ws_size >= need);
    float* expw = (float*)d_ws;

    // Kernel 1: affinity + exp precompute
    if (has_ws)
        aff_kernel<true><<<dim3(C_), 256, 0, stream>>>(x, keys, aff, expw);
    else
        aff_kernel<false><<<dim3(C_), 256, 0, stream>>>(x, keys, aff, expw);

    // Zero out-accumulator (split-K atomics), then kernel 2.
    hipMemsetAsync(out, 0, (size_t)N_ * C_ * sizeof(float), stream);
    if (has_ws)
        out_kernel<false><<<dim3(32, CK_ / KCHUNK), 256, 0, stream>>>(expw, values, out);
    else
        out_kernel<true><<<dim3(32, CK_ / KCHUNK), 256, 0, stream>>>(aff, values, out);
}